// GloVeGAT_39273180955003
// MI455X (gfx1250) — compile-verified
//
#include <hip/hip_runtime.h>
#include <hip/hip_bf16.h>
#include <math.h>

// ---------------- problem constants ----------------
#define NN    8192         // nodes / LSTM sequence length
#define EE    262144       // edges before self loops
#define ETOT  (EE + NN)    // edges incl self loops = 270336
#define EMB   300
#define HH    8            // heads
#define CC    100          // per-head dim / lstm hidden
#define HC    800          // HH*CC
#define LL    8            // lstm layers
#define G4C   400          // 4*CC gates
#define NQ    32

typedef float v2f __attribute__((ext_vector_type(2)));
typedef float v8f __attribute__((ext_vector_type(8)));

// ---------------- workspace layout (float offsets) ----------------
#define OFF_X     ((size_t)0)                        // [NN*HC]  lstm input (x0 / gat out)
#define OFF_XW    (OFF_X   + (size_t)NN*HC)          // [NN*HC]  gat xw
#define OFF_ACC   (OFF_XW  + (size_t)NN*HC)          // [NN*HC]  gat scatter accum
#define OFF_Y     (OFF_ACC + (size_t)NN*HC)          // [LL*NN*CC] inter-layer lstm outputs
#define OFF_ES    (OFF_Y   + (size_t)LL*NN*CC)       // [NN*HH]
#define OFF_ED    (OFF_ES  + (size_t)NN*HH)          // [NN*HH]
#define OFF_DEN   (OFF_ED  + (size_t)NN*HH)          // [NN*HH]
#define OFF_EX    (OFF_DEN + (size_t)NN*HH)          // [ETOT*HH]
#define OFF_T0    (OFF_EX  + (size_t)ETOT*HH)        // [HC*G4C]   W_ih0^T
#define OFF_TR    (OFF_T0  + (size_t)HC*G4C)         // [7*CC*G4C] W_ih_rest^T
#define OFF_TH    (OFF_TR  + (size_t)7*CC*G4C)       // [8*CC*G4C] W_hh^T
#define OFF_HS    (OFF_TH  + (size_t)8*CC*G4C)       // [LL*CC]
#define OFF_CS    (OFF_HS  + (size_t)LL*CC)          // [LL*CC]
#define OFF_FLAGS (OFF_CS  + (size_t)LL*CC)          // [LL*NN] ints

// ==================================================================
// GEMM with V_WMMA_F32_16X16X4_F32:  C[M,Nn] = act(A[M,K] @ B + bias)
// B element (k,n) read at  B + n*ldbn + k   (i.e. B given transposed-major)
// Optional row gather on A (fused embedding lookup).
// One wave (32 lanes) computes one 16x16 tile.
// ==================================================================
__global__ __launch_bounds__(32)
void gemm_wmma_f32(const float* __restrict__ A, const float* __restrict__ B,
                   const float* __restrict__ bias, float* __restrict__ Cmat,
                   int M, int Nn, int K, int lda, int ldbn, int ldc,
                   int relu, const int* __restrict__ gather)
{
    const int tile_n = blockIdx.x * 16;
    const int tile_m = blockIdx.y * 16;
    const int lane   = threadIdx.x;          // 0..31, wave32
    const int half   = lane >> 4;            // 0: K=k0,k0+1   1: K=k0+2,k0+3
    const int lr     = lane & 15;

    const int arow = tile_m + lr;
    const float* Arow = gather ? (A + (size_t)gather[arow] * lda)
                               : (A + (size_t)arow * lda);
    const float* Bcol = B + (size_t)(tile_n + lr) * ldbn;

    v8f acc = {};
    for (int k0 = 0; k0 < K; k0 += 4) {
        const int ka = k0 + half * 2;
        __builtin_prefetch(Arow + ka + 32, 0, 1);
        __builtin_prefetch(Bcol + ka + 32, 0, 1);
        v2f a = *(const v2f*)(Arow + ka);    // A frag: 16x4 f32, 2 VGPR/lane
        v2f b = *(const v2f*)(Bcol + ka);    // B frag: 4x16 f32, 2 VGPR/lane
        acc = __builtin_amdgcn_wmma_f32_16x16x4_f32(
                  false, a, false, b, (short)0, acc, false, false);
    }

    const int col   = tile_n + lr;
    const float bv  = bias ? bias[col] : 0.0f;
    const int mbase = tile_m + half * 8;     // C/D layout: vgpr r -> M=r(+8)
#pragma unroll
    for (int r = 0; r < 8; ++r) {
        float v = acc[r] + bv;
        if (relu) v = v > 0.0f ? v : 0.0f;
        Cmat[(size_t)(mbase + r) * ldc + col] = v;
    }
}

// ==================================================================
// transpose LSTM weights once per launch -> coalesced matvec reads
// ==================================================================
__global__ void transpose_lstm_weights(const float* __restrict__ Wih0,
                                       const float* __restrict__ Wihr,
                                       const float* __restrict__ Whh,
                                       float* __restrict__ T0,
                                       float* __restrict__ Tr,
                                       float* __restrict__ Th)
{
    const int i = blockIdx.x * blockDim.x + threadIdx.x;
    if (i < HC * G4C) {                       // T0[k][j] = Wih0[j][k]
        int k = i / G4C, j = i % G4C;
        T0[i] = Wih0[(size_t)j * HC + k];
    }
    if (i < 7 * CC * G4C) {                   // Tr[l][k][j] = Wihr[l][j][k]
        int l = i / (CC * G4C), rem = i % (CC * G4C);
        int k = rem / G4C, j = rem % G4C;
        Tr[i] = Wihr[(size_t)l * G4C * CC + (size_t)j * CC + k];
    }
    if (i < 8 * CC * G4C) {                   // Th[l][k][j] = Whh[l][j][k]
        int l = i / (CC * G4C), rem = i % (CC * G4C);
        int k = rem / G4C, j = rem % G4C;
        Th[i] = Whh[(size_t)l * G4C * CC + (size_t)j * CC + k];
    }
}

// ==================================================================
// 8-layer LSTM scan, systolically pipelined: block l == layer l,
// consumes layer l-1's per-timestep output via acquire/release flags.
// ==================================================================
__global__ __launch_bounds__(512)
void lstm_scan(const float* __restrict__ x,        // [NN, HC] (layer-0 input)
               float* __restrict__ ybuf,           // [LL, NN, CC]
               float* __restrict__ hstate,         // [LL, CC]
               float* __restrict__ cstate,         // [LL, CC]
               const float* __restrict__ T0,
               const float* __restrict__ Tr,
               const float* __restrict__ Th,
               const float* __restrict__ bih,      // [LL, 4C]
               const float* __restrict__ bhh,      // [LL, 4C]
               int* __restrict__ flags)            // [LL, NN], zeroed
{
    __shared__ float s_in[HC];
    __shared__ float s_h[CC];
    __shared__ float s_c[CC];
    __shared__ float s_g[G4C];

    const int l   = blockIdx.x;
    const int tid = threadIdx.x;
    const int Din = (l == 0) ? HC : CC;
    const float* WiT = (l == 0) ? T0 : (Tr + (size_t)(l - 1) * CC * G4C);
    const float* WhT = Th + (size_t)l * CC * G4C;

    if (tid < CC) { s_h[tid] = hstate[l * CC + tid]; s_c[tid] = cstate[l * CC + tid]; }
    const float bsum = (tid < G4C) ? (bih[l * G4C + tid] + bhh[l * G4C + tid]) : 0.0f;
    __syncthreads();

    for (int t = 0; t < NN; ++t) {
        // --- acquire input vector ---
        if (l > 0) {
            if (tid == 0) {
                while (__hip_atomic_load(&flags[(l - 1) * NN + t],
                                         __ATOMIC_ACQUIRE,
                                         __HIP_MEMORY_SCOPE_AGENT) == 0) {
                    __builtin_amdgcn_s_sleep(1);
                }
            }
            __syncthreads();
            for (int k = tid; k < Din; k += 512)
                s_in[k] = ybuf[((size_t)(l - 1) * NN + t) * CC + k];
        } else {
            for (int k = tid; k < Din; k += 512)
                s_in[k] = x[(size_t)t * HC + k];
        }
        __syncthreads();

        // --- gates: g = WiT^T*in + WhT^T*h + b  (coalesced weight reads) ---
        if (tid < G4C) {
            float g = bsum;
            const float* wp = WiT + tid;
            for (int k = 0; k < Din; k += 4) {
                g += wp[(size_t)(k    ) * G4C] * s_in[k    ];
                g += wp[(size_t)(k + 1) * G4C] * s_in[k + 1];
                g += wp[(size_t)(k + 2) * G4C] * s_in[k + 2];
                g += wp[(size_t)(k + 3) * G4C] * s_in[k + 3];
            }
            const float* hp = WhT + tid;
            for (int k = 0; k < CC; k += 4) {
                g += hp[(size_t)(k    ) * G4C] * s_h[k    ];
                g += hp[(size_t)(k + 1) * G4C] * s_h[k + 1];
                g += hp[(size_t)(k + 2) * G4C] * s_h[k + 2];
                g += hp[(size_t)(k + 3) * G4C] * s_h[k + 3];
            }
            s_g[tid] = g;
        }
        __syncthreads();

        // --- cell update (gate order i,f,g,o), publish h ---
        if (tid < CC) {
            float ig = 1.0f / (1.0f + expf(-s_g[tid]));
            float fg = 1.0f / (1.0f + expf(-s_g[CC + tid]));
            float gg = tanhf(s_g[2 * CC + tid]);
            float og = 1.0f / (1.0f + expf(-s_g[3 * CC + tid]));
            float c  = fg * s_c[tid] + ig * gg;
            float h  = og * tanhf(c);
            s_c[tid] = c;
            s_h[tid] = h;
            ybuf[((size_t)l * NN + t) * CC + tid] = h;
            __threadfence();
        }
        __syncthreads();
        if (tid == 0)
            __hip_atomic_store(&flags[l * NN + t], 1,
                               __ATOMIC_RELEASE, __HIP_MEMORY_SCOPE_AGENT);
        __syncthreads();
    }

    if (tid < CC) { hstate[l * CC + tid] = s_h[tid]; cstate[l * CC + tid] = s_c[tid]; }
}

// ==================================================================
// GAT pieces
// ==================================================================
__global__ void att_scores(const float* __restrict__ xw,
                           const float* __restrict__ as_, const float* __restrict__ ad_,
                           float* __restrict__ es, float* __restrict__ ed)
{
    const int i = blockIdx.x * blockDim.x + threadIdx.x;   // n*HH + h
    if (i >= NN * HH) return;
    const int n = i >> 3, h = i & 7;
    const float* xp = xw + (size_t)n * HC + h * CC;
    float s = 0.0f, d = 0.0f;
    for (int c = 0; c < CC; ++c) { float v = xp[c]; s += v * as_[h * CC + c]; d += v * ad_[h * CC + c]; }
    es[i] = s; ed[i] = d;
}

// exp(leakyrelu(es[src]+ed[dst])) per (edge,head); accumulate denom at dst.
// (segment-max subtraction cancels exactly in alpha; values are O(1), safe.)
__global__ void edge_softmax_num(const int* __restrict__ ei,
                                 const float* __restrict__ es, const float* __restrict__ ed,
                                 float* __restrict__ ex, float* __restrict__ den)
{
    const int i = blockIdx.x * blockDim.x + threadIdx.x;   // e*HH + h
    if (i >= ETOT * HH) return;
    const int e = i >> 3, h = i & 7;
    int s, d;
    if (e < EE) { s = ei[e]; d = ei[EE + e]; } else { s = d = e - EE; }
    float v = es[s * HH + h] + ed[d * HH + h];
    v = v > 0.0f ? v : 0.2f * v;                          // LeakyReLU(0.2)
    float xv = expf(v);
    ex[i] = xv;
    atomicAdd(&den[d * HH + h], xv);
}

// scatter:  acc[dst, h, c] += ex[e,h] * xw[src, h, c]
__global__ __launch_bounds__(256)
void edge_scatter(const int* __restrict__ ei, const float* __restrict__ ex,
                  const float* __restrict__ xw, float* __restrict__ acc)
{
    const int e = blockIdx.x;
    int s, d;
    if (e < EE) { s = ei[e]; d = ei[EE + e]; } else { s = d = e - EE; }
    const float* xs = xw + (size_t)s * HC;
    float* ap = acc + (size_t)d * HC;
    for (int hc = threadIdx.x; hc < HC; hc += blockDim.x) {
        const float a = ex[(size_t)e * HH + (hc / CC)];
        atomicAdd(&ap[hc], a * xs[hc]);
    }
}

// g = relu(acc/denom + gat_b)  -> next LSTM input
__global__ void gat_finalize(const float* __restrict__ acc, const float* __restrict__ den,
                             const float* __restrict__ gb, float* __restrict__ outx)
{
    const size_t i = (size_t)blockIdx.x * blockDim.x + threadIdx.x;
    if (i >= (size_t)NN * HC) return;
    const int n = (int)(i / HC), hc = (int)(i % HC), h = hc / CC;
    float v = acc[i] / den[n * HH + h] + gb[hc];
    outx[i] = v > 0.0f ? v : 0.0f;
}

// ==================================================================
// pooled head:  (mean over queries) @ lin_W^T + lin_b -> d_out[3]
// ==================================================================
__global__ __launch_bounds__(128)
void head_kernel(const float* __restrict__ out, const int* __restrict__ q,
                 const float* __restrict__ linW, const float* __restrict__ linb,
                 float* __restrict__ res)
{
    __shared__ float pooled[CC];
    const int tid = threadIdx.x;
    if (tid < CC) {
        float s = 0.0f;
        for (int i = 0; i < NQ; ++i) s += out[(size_t)q[i] * CC + tid];
        pooled[tid] = s * (1.0f / NQ);
    }
    __syncthreads();
    if (tid < 3) {
        float s = linb[tid];
        for (int c = 0; c < CC; ++c) s += linW[tid * CC + c] * pooled[c];
        res[tid] = s;
    }
}

// ==================================================================
extern "C" void kernel_launch(void* const* d_in, const int* in_sizes, int n_in,
                              void* d_out, int out_size, void* d_ws, size_t ws_size,
                              hipStream_t stream)
{
    (void)in_sizes; (void)n_in; (void)out_size; (void)ws_size;

    const int*   node_ids = (const int*)  d_in[0];
    const int*   edge_idx = (const int*)  d_in[1];
    const int*   q_idx    = (const int*)  d_in[2];
    const float* emb      = (const float*)d_in[3];
    const float* W_init   = (const float*)d_in[4];
    const float* b_init   = (const float*)d_in[5];
    const float* gat_W    = (const float*)d_in[6];
    const float* att_src  = (const float*)d_in[7];
    const float* att_dst  = (const float*)d_in[8];
    const float* gat_b    = (const float*)d_in[9];
    const float* W_ih0    = (const float*)d_in[10];
    const float* W_ihr    = (const float*)d_in[11];
    const float* W_hh     = (const float*)d_in[12];
    const float* b_ih     = (const float*)d_in[13];
    const float* b_hh     = (const float*)d_in[14];
    const float* lin_W    = (const float*)d_in[15];
    const float* lin_b    = (const float*)d_in[16];

    float* ws   = (float*)d_ws;
    float* X    = ws + OFF_X;
    float* XW   = ws + OFF_XW;
    float* ACC  = ws + OFF_ACC;
    float* Y    = ws + OFF_Y;
    float* ES   = ws + OFF_ES;
    float* ED   = ws + OFF_ED;
    float* DEN  = ws + OFF_DEN;
    float* EX   = ws + OFF_EX;
    float* T0   = ws + OFF_T0;
    float* TR   = ws + OFF_TR;
    float* TH   = ws + OFF_TH;
    float* HS   = ws + OFF_HS;
    float* CS   = ws + OFF_CS;
    int*   FLG  = (int*)(ws + OFF_FLAGS);
    float* OUTL = Y + (size_t)(LL - 1) * NN * CC;   // last-layer hidden = lstm out

    // fresh recurrent state each launch
    hipMemsetAsync(HS, 0, sizeof(float) * 2 * LL * CC, stream);

    // pre-transpose LSTM weights (coalesced matvec reads)
    transpose_lstm_weights<<<(HC * G4C + 255) / 256, 256, 0, stream>>>(
        W_ih0, W_ihr, W_hh, T0, TR, TH);

    // x0 = relu(emb[node_ids] @ W_init^T + b_init)   [WMMA f32]
    gemm_wmma_f32<<<dim3(HC / 16, NN / 16), 32, 0, stream>>>(
        emb, W_init, b_init, X, NN, HC, EMB, EMB, EMB, HC, 1, node_ids);

    // LSTM pass 1
    hipMemsetAsync(FLG, 0, sizeof(int) * LL * NN, stream);
    lstm_scan<<<LL, 512, 0, stream>>>(X, Y, HS, CS, T0, TR, TH, b_ih, b_hh, FLG);

    for (int k = 0; k < 3; ++k) {
        // xw = out @ gat_W[k]^T   [8192x100x800, WMMA f32]
        gemm_wmma_f32<<<dim3(HC / 16, NN / 16), 32, 0, stream>>>(
            OUTL, gat_W + (size_t)k * HC * CC, nullptr, XW,
            NN, HC, CC, CC, CC, HC, 0, nullptr);

        att_scores<<<(NN * HH + 255) / 256, 256, 0, stream>>>(
            XW, att_src + k * HH * CC, att_dst + k * HH * CC, ES, ED);

        hipMemsetAsync(DEN, 0, sizeof(float) * NN * HH, stream);
        hipMemsetAsync(ACC, 0, sizeof(float) * (size_t)NN * HC, stream);

        edge_softmax_num<<<(ETOT * HH + 255) / 256, 256, 0, stream>>>(
            edge_idx, ES, ED, EX, DEN);

        edge_scatter<<<ETOT, 256, 0, stream>>>(edge_idx, EX, XW, ACC);

        gat_finalize<<<(int)(((size_t)NN * HC + 255) / 256), 256, 0, stream>>>(
            ACC, DEN, gat_b + k * HC, X);

        // LSTM pass k+2 (state carried in HS/CS)
        hipMemsetAsync(FLG, 0, sizeof(int) * LL * NN, stream);
        lstm_scan<<<LL, 512, 0, stream>>>(X, Y, HS, CS, T0, TR, TH, b_ih, b_hh, FLG);
    }

    head_kernel<<<1, 128, 0, stream>>>(OUTL, q_idx, lin_W, lin_b, (float*)d_out);
}